// SAGE_52381421142170
// MI455X (gfx1250) — compile-verified
//
#include <hip/hip_runtime.h>

// GraphSAGE 3-layer forward for MI455X (gfx1250).
// Aggregation (edge gather + float atomic scatter) is the memory-bound part;
// GEMMs use exact-precision V_WMMA_F32_16X16X4_F32 (wave32, 16x16 f32 tile).

#define N_NODES 100000
#define N_EDGES 1600000
#define IN_F 128
#define HID_F 256
#define OUT_F 64

typedef __attribute__((ext_vector_type(2))) float v2f;
typedef __attribute__((ext_vector_type(8))) float v8f;

// ---------------- utility kernels ----------------

__global__ void sage_fill_zero(float* __restrict__ p, long long n) {
  long long i = (long long)blockIdx.x * blockDim.x + threadIdx.x;
  if (i < n) p[i] = 0.0f;
}

__global__ void sage_degree(const int* __restrict__ dst, float* __restrict__ deg, int E) {
  int i = blockIdx.x * blockDim.x + threadIdx.x;
  if (i < E) atomicAdd(&deg[dst[i]], 1.0f);
}

__global__ void sage_rdeg(float* __restrict__ deg, int n) {
  int i = blockIdx.x * blockDim.x + threadIdx.x;
  if (i < n) deg[i] = 1.0f / fmaxf(deg[i], 1.0f);
}

// ---------------- edge aggregation (memory-bound part) ----------------
// One thread per (edge, 4-float chunk). Consecutive threads cover consecutive
// chunks of one edge's source row -> coalesced 16B reads; scatter via
// global_atomic_add_f32 (no return -> STOREcnt, no wait needed).

__global__ void sage_aggregate(const float* __restrict__ h,
                               const int* __restrict__ src,
                               const int* __restrict__ dst,
                               float* __restrict__ agg,
                               int E, int d) {
  const int cpe = d >> 2;  // float4 chunks per edge row
  long long gid = (long long)blockIdx.x * blockDim.x + threadIdx.x;
  long long total = (long long)E * cpe;
  if (gid >= total) return;
  int e = (int)(gid / cpe);
  int c = (int)(gid % cpe);
  int s = src[e];
  int t = dst[e];
  const float4 v = *((const float4*)(h + (long long)s * d) + c);
  float* ap = agg + (long long)t * d + (c << 2);
  atomicAdd(ap + 0, v.x);
  atomicAdd(ap + 1, v.y);
  atomicAdd(ap + 2, v.z);
  atomicAdd(ap + 3, v.w);
}

// ---------------- fused SAGE GEMM with f32 WMMA ----------------
// OUT[M=nodes, NO] = H @ Wself^T + (AGG*rdeg) @ Wneigh^T + b  (opt. ReLU)
// One wave32 per 16x16 output tile; K stepped by 4 with V_WMMA_F32_16X16X4_F32.
//
// ISA VGPR layouts (CDNA5 ISA):
//   A (16x4 f32):  lane m=lane&15, VGPR v holds K = kk + (lane>=16)*2 + v
//   B (4x16 f32):  lane n=lane&15, VGPR v holds K = kk + (lane>=16)*2 + v
//   C/D (16x16):   VGPR r: lanes0-15 -> M=r, lanes16-31 -> M=r+8; N=lane&15

template <bool RELU>
__global__ void sage_gemm_wmma(const float* __restrict__ H,
                               const float* __restrict__ AGG,
                               const float* __restrict__ rdeg,
                               const float* __restrict__ Wself,
                               const float* __restrict__ Wneigh,
                               const float* __restrict__ bias,
                               float* __restrict__ OUT,
                               int K, int NO) {
  const int lane = threadIdx.x & 31;
  const int wave = threadIdx.x >> 5;
  const int m_base = (blockIdx.x * 8 + wave) * 16;
  if (m_base >= N_NODES) return;  // wave-uniform: EXEC stays all-ones
  const int n_base = blockIdx.y * 16;

  const int m = lane & 15;            // A row within tile / D column selector
  const int klo = (lane >> 4) << 1;   // K sub-offset for this half-wave
  const int row = m_base + m;         // node index for A loads
  const int col = n_base + m;         // output-feature index for B loads

  const float* __restrict__ hrow = H + (long long)row * K;
  const float* __restrict__ arow = AGG + (long long)row * K;
  const float* __restrict__ wsr = Wself + (long long)col * K;   // B[k][n]=W[n][k]
  const float* __restrict__ wnr = Wneigh + (long long)col * K;
  const float rd = rdeg[row];  // 1/max(deg,1), folded into neighbor A operand

  v8f acc = {};
  for (int kk = 0; kk < K; kk += 4) {
    v2f a_s = *(const v2f*)(hrow + kk + klo);
    v2f b_s = *(const v2f*)(wsr + kk + klo);
    v2f a_n = *(const v2f*)(arow + kk + klo);
    v2f b_n = *(const v2f*)(wnr + kk + klo);
    a_n *= rd;
    acc = __builtin_amdgcn_wmma_f32_16x16x4_f32(
        false, a_s, false, b_s, (short)0, acc, false, false);
    acc = __builtin_amdgcn_wmma_f32_16x16x4_f32(
        false, a_n, false, b_n, (short)0, acc, false, false);
  }

  const float bj = bias[n_base + (lane & 15)];
  const int mrow0 = m_base + ((lane >> 4) << 3);  // +8 for lanes 16..31
  const int j = n_base + (lane & 15);
#pragma unroll
  for (int r = 0; r < 8; ++r) {
    float v = acc[r] + bj;
    if (RELU) v = fmaxf(v, 0.0f);
    OUT[(long long)(mrow0 + r) * NO + j] = v;  // lanes 0-15 / 16-31 each contiguous
  }
}

// ---------------- driver ----------------

static inline int cdiv(long long a, long long b) { return (int)((a + b - 1) / b); }

extern "C" void kernel_launch(void* const* d_in, const int* in_sizes, int n_in,
                              void* d_out, int out_size, void* d_ws, size_t ws_size,
                              hipStream_t stream) {
  const float* feats = (const float*)d_in[0];
  const int* esrc = (const int*)d_in[1];
  const int* edst = (const int*)d_in[2];
  const float* W0s = (const float*)d_in[3];
  const float* W0n = (const float*)d_in[4];
  const float* b0 = (const float*)d_in[5];
  const float* W1s = (const float*)d_in[6];
  const float* W1n = (const float*)d_in[7];
  const float* b1 = (const float*)d_in[8];
  const float* W2s = (const float*)d_in[9];
  const float* W2n = (const float*)d_in[10];
  const float* b2 = (const float*)d_in[11];
  float* out = (float*)d_out;

  const long long NH = (long long)N_NODES * HID_F;
  float* H1 = (float*)d_ws;          // [N, 256]
  float* H2 = H1 + NH;               // [N, 256]
  float* AGG = H2 + NH;              // [N, 256] (layer0 uses first N*128)
  float* DEG = AGG + NH;             // [N] -> becomes 1/max(deg,1)

  const int TB = 256;
  const dim3 blk(TB);

  // --- degree (shared across layers) ---
  sage_fill_zero<<<cdiv(N_NODES, TB), blk, 0, stream>>>(DEG, N_NODES);
  sage_degree<<<cdiv(N_EDGES, TB), blk, 0, stream>>>(edst, DEG, N_EDGES);
  sage_rdeg<<<cdiv(N_NODES, TB), blk, 0, stream>>>(DEG, N_NODES);

  const int m_blocks = cdiv(N_NODES, 16 * 8);  // 8 waves (tiles) per block

  // --- layer 0: IN_F -> HID_F, ReLU ---
  sage_fill_zero<<<cdiv((long long)N_NODES * IN_F, TB), blk, 0, stream>>>(
      AGG, (long long)N_NODES * IN_F);
  sage_aggregate<<<cdiv((long long)N_EDGES * (IN_F / 4), TB), blk, 0, stream>>>(
      feats, esrc, edst, AGG, N_EDGES, IN_F);
  sage_gemm_wmma<true><<<dim3(m_blocks, HID_F / 16), blk, 0, stream>>>(
      feats, AGG, DEG, W0s, W0n, b0, H1, IN_F, HID_F);

  // --- layer 1: HID_F -> HID_F, ReLU ---
  sage_fill_zero<<<cdiv(NH, TB), blk, 0, stream>>>(AGG, NH);
  sage_aggregate<<<cdiv((long long)N_EDGES * (HID_F / 4), TB), blk, 0, stream>>>(
      H1, esrc, edst, AGG, N_EDGES, HID_F);
  sage_gemm_wmma<true><<<dim3(m_blocks, HID_F / 16), blk, 0, stream>>>(
      H1, AGG, DEG, W1s, W1n, b1, H2, HID_F, HID_F);

  // --- layer 2: HID_F -> OUT_F, no activation ---
  sage_fill_zero<<<cdiv(NH, TB), blk, 0, stream>>>(AGG, NH);
  sage_aggregate<<<cdiv((long long)N_EDGES * (HID_F / 4), TB), blk, 0, stream>>>(
      H2, esrc, edst, AGG, N_EDGES, HID_F);
  sage_gemm_wmma<false><<<dim3(m_blocks, OUT_F / 16), blk, 0, stream>>>(
      H2, AGG, DEG, W2s, W2n, b2, out, HID_F, OUT_F);
}